// SpaceSelfAttentionModality_66709432041573
// MI455X (gfx1250) — compile-verified
//
#include <hip/hip_runtime.h>

typedef float v2f __attribute__((ext_vector_type(2)));
typedef float v8f __attribute__((ext_vector_type(8)));

#define S_LEN 512
#define D_MODEL 768
#define D3 2304
#define NHEAD 12
#define HDIM 64
#define NSEQ 32

static constexpr size_t QKVSZ = (size_t)NSEQ * NHEAD * S_LEN * HDIM; // 12,582,912 floats

// D = A(16x4 f32) * B(4x16 f32) + C(16x16 f32), wave32
__device__ __forceinline__ v8f wmma4(v2f a, v2f b, v8f c) {
    return __builtin_amdgcn_wmma_f32_16x16x4_f32(
        /*neg_a=*/false, a, /*neg_b=*/false, b,
        /*c_mod=*/(short)0, c, /*reuse_a=*/false, /*reuse_b=*/false);
}

// CDNA5 async DMA: global -> LDS, 16B per lane, tracked by ASYNCcnt.
__device__ __forceinline__ void async_load_b128(unsigned lds_off, const float* gaddr) {
    asm volatile("global_load_async_to_lds_b128 %0, %1, off"
                 :: "v"(lds_off), "v"(gaddr) : "memory");
}
__device__ __forceinline__ void wait_async0() {
    asm volatile("s_wait_asynccnt 0x0" ::: "memory");
}
__device__ __forceinline__ unsigned lds_off_of(const void* p) {
    return (unsigned)(unsigned long long)p;   // low 32 bits of flat LDS addr = WG offset
}

// ---------------------------------------------------------------------------
// Kernel 1: qkv = x @ w_qkv + b_qkv, scattered to Q/K/V [n][h][s][64] in ws.
// grid (36, 4, 32), block 256. Block tile 128(M) x 64(N); wave tile 16 x 64.
// W tile (16K x 64N) staged via async-to-LDS, double buffered, stride 72.
// ---------------------------------------------------------------------------
__global__ __launch_bounds__(256)
void qkv_gemm_kernel(const float* __restrict__ x, const float* __restrict__ w,
                     const float* __restrict__ bias, float* __restrict__ ws) {
    __shared__ float wbuf[2][16 * 72];

    const int tid  = threadIdx.x;
    const int wave = tid >> 5;
    const int lane = tid & 31;
    const int lr   = lane & 15;
    const int half = lane >> 4;

    const int n    = blockIdx.z;
    const int m0   = blockIdx.y * 128 + wave * 16;
    const int col0 = blockIdx.x * 64;

    const int srow_ld = tid >> 4;          // staging: row within 16-K tile
    const int scol_ld = (tid & 15) * 4;    // staging: 4-dword column group

    const float* A = x + (size_t)n * S_LEN * D_MODEL + (size_t)(m0 + lr) * D_MODEL + 2 * half;

    // prologue: stage K-tile 0
    async_load_b128(lds_off_of(&wbuf[0][srow_ld * 72 + scol_ld]),
                    w + (size_t)srow_ld * D3 + col0 + scol_ld);

    v8f acc[4] = {};
    const int NK = D_MODEL / 16;           // 48
    for (int kt = 0; kt < NK; ++kt) {
        wait_async0();                     // own async ops for tile kt done
        __syncthreads();                   // tile kt visible; prev buffer free
        if (kt + 1 < NK) {
            async_load_b128(lds_off_of(&wbuf[(kt + 1) & 1][srow_ld * 72 + scol_ld]),
                            w + (size_t)((kt + 1) * 16 + srow_ld) * D3 + col0 + scol_ld);
        }
        const float* wt = &wbuf[kt & 1][0];
#pragma unroll
        for (int k0s = 0; k0s < 4; ++k0s) {
            v2f a = *(const v2f*)(A + kt * 16 + k0s * 4);
            int krow = k0s * 4 + 2 * half;
#pragma unroll
            for (int t = 0; t < 4; ++t) {
                v2f b;
                b.x = wt[krow * 72 + t * 16 + lr];
                b.y = wt[(krow + 1) * 72 + t * 16 + lr];
                acc[t] = wmma4(a, b, acc[t]);
            }
        }
    }

#pragma unroll
    for (int t = 0; t < 4; ++t) {
        int col = col0 + t * 16 + lr;      // [0, 2304)
        int p   = col / 768;               // 0=q 1=k 2=v (uniform per instr)
        int hh  = (col - p * 768) >> 6;    // head
        int dh  = col & 63;                // dim within head
        float bv = bias[col];
        float* dst = ws + (size_t)p * QKVSZ
                        + ((size_t)(n * NHEAD + hh) * S_LEN) * HDIM + dh;
#pragma unroll
        for (int r = 0; r < 8; ++r) {
            int srow = m0 + r + 8 * half;
            dst[(size_t)srow * HDIM] = acc[t][r] + bv;
        }
    }
}

// ---------------------------------------------------------------------------
// Kernel 2: masked flash attention per (n, head). Each wave owns 16 q-rows.
// grid (4 qblk, 12 h, 32 n), block 256 (8 waves -> 128 q-rows per block).
// K/V tiles (16x64) staged block-wide via async-to-LDS, double buffered.
// K stride 68 (lane-indexed rows), V stride 72 (fixed-row reads).
// ---------------------------------------------------------------------------
__global__ __launch_bounds__(256)
void attn_kernel(const float* __restrict__ ws, const unsigned char* __restrict__ mask,
                 float* __restrict__ y) {
    __shared__ float kbuf[2][16 * 68];
    __shared__ float vbuf[2][16 * 72];
    __shared__ float plds[8][16 * 18];     // per-wave padded 16x16 P tile

    const int tid  = threadIdx.x;
    const int wave = tid >> 5;
    const int lane = tid & 31;
    const int lr   = lane & 15;
    const int half = lane >> 4;

    const int h  = blockIdx.y;
    const int n  = blockIdx.z;
    const int q0 = blockIdx.x * 128 + wave * 16;

    const size_t head_off = ((size_t)(n * NHEAD + h) * S_LEN) * HDIM;
    const float* qp = ws + head_off;
    const float* kp = ws + QKVSZ + head_off;
    const float* vp = ws + 2 * QKVSZ + head_off;
    float* pl = &plds[wave][0];

    const int srow_ld = tid >> 4;
    const int scol_ld = (tid & 15) * 4;

    // Preload Q A-frags: a.x = Q[q0+lr][4*ks + 2*half], a.y = next k
    v2f aq[16];
    const float* ql = qp + (size_t)(q0 + lr) * HDIM + 2 * half;
#pragma unroll
    for (int ks = 0; ks < 16; ++ks) aq[ks] = *(const v2f*)(ql + 4 * ks);

    v8f acc[4] = {};
    float m_run[8], l_run[8];
#pragma unroll
    for (int r = 0; r < 8; ++r) { m_run[r] = -__builtin_inff(); l_run[r] = 0.0f; }

    // prologue: stage K/V tile 0
    async_load_b128(lds_off_of(&kbuf[0][srow_ld * 68 + scol_ld]),
                    kp + (size_t)srow_ld * HDIM + scol_ld);
    async_load_b128(lds_off_of(&vbuf[0][srow_ld * 72 + scol_ld]),
                    vp + (size_t)srow_ld * HDIM + scol_ld);

    const int NKB = S_LEN / 16;            // 32
    for (int kbi = 0; kbi < NKB; ++kbi) {
        wait_async0();
        __syncthreads();
        if (kbi + 1 < NKB) {
            int slot = (kbi + 1) & 1;
            async_load_b128(lds_off_of(&kbuf[slot][srow_ld * 68 + scol_ld]),
                            kp + (size_t)((kbi + 1) * 16 + srow_ld) * HDIM + scol_ld);
            async_load_b128(lds_off_of(&vbuf[slot][srow_ld * 72 + scol_ld]),
                            vp + (size_t)((kbi + 1) * 16 + srow_ld) * HDIM + scol_ld);
        }
        const float* kt_ = &kbuf[kbi & 1][0];
        const float* vt_ = &vbuf[kbi & 1][0];
        const int kb = kbi * 16;

        // S = Q @ K^T for this 16x16 tile (B-frag row = key index lr)
        v8f S = {};
#pragma unroll
        for (int ks = 0; ks < 16; ++ks) {
            v2f bk = *(const v2f*)(kt_ + lr * 68 + 4 * ks + 2 * half);
            S = wmma4(aq[ks], bk, S);
        }

        // mask + online softmax (row = r + 8*half lives across 16 lanes)
        float p[8], alpha[8];
#pragma unroll
        for (int r = 0; r < 8; ++r) {
            int qrow = q0 + r + 8 * half;
            bool ok  = mask[(size_t)qrow * S_LEN + kb + lr] != 0;
            float sv = ok ? S[r] * 0.125f : -__builtin_inff();
            float mx = sv;
            mx = fmaxf(mx, __shfl_xor(mx, 1, 16));
            mx = fmaxf(mx, __shfl_xor(mx, 2, 16));
            mx = fmaxf(mx, __shfl_xor(mx, 4, 16));
            mx = fmaxf(mx, __shfl_xor(mx, 8, 16));
            float mnew  = fmaxf(m_run[r], mx);
            float msafe = (mnew == -__builtin_inff()) ? 0.0f : mnew;
            float a_ = __expf(m_run[r] - msafe);   // -inf -> 0, harmless (l=0)
            float pv = __expf(sv - msafe);         // masked -> 0
            float ps = pv;
            ps += __shfl_xor(ps, 1, 16);
            ps += __shfl_xor(ps, 2, 16);
            ps += __shfl_xor(ps, 4, 16);
            ps += __shfl_xor(ps, 8, 16);
            l_run[r] = l_run[r] * a_ + ps;
            m_run[r] = mnew;
            alpha[r] = a_;
            p[r]     = pv;
        }
#pragma unroll
        for (int t = 0; t < 4; ++t)
#pragma unroll
            for (int r = 0; r < 8; ++r) acc[t][r] *= alpha[r];

        // C-layout -> A-layout of P via per-wave LDS (stride 18, 8B aligned).
        // DS pipe is in-order per wave; fences stop compiler reordering.
#pragma unroll
        for (int r = 0; r < 8; ++r) pl[(r + 8 * half) * 18 + lr] = p[r];
        asm volatile("" ::: "memory");
        v2f pa[4];
#pragma unroll
        for (int ks2 = 0; ks2 < 4; ++ks2)
            pa[ks2] = *(const v2f*)(pl + lr * 18 + 4 * ks2 + 2 * half);
        asm volatile("" ::: "memory");

        // acc += P @ V (V B-frags from LDS tile)
#pragma unroll
        for (int t = 0; t < 4; ++t) {
#pragma unroll
            for (int ks2 = 0; ks2 < 4; ++ks2) {
                int kr = 4 * ks2 + 2 * half;
                v2f bv;
                bv.x = vt_[kr * 72 + t * 16 + lr];
                bv.y = vt_[(kr + 1) * 72 + t * 16 + lr];
                acc[t] = wmma4(pa[ks2], bv, acc[t]);
            }
        }
    }

    // y[n][s][h*64 + dh] = acc / l
    float* yb = y + (size_t)n * S_LEN * D_MODEL + h * HDIM;
#pragma unroll
    for (int r = 0; r < 8; ++r) {
        float rl = 1.0f / l_run[r];
        int qrow = q0 + r + 8 * half;
#pragma unroll
        for (int t = 0; t < 4; ++t)
            yb[(size_t)qrow * D_MODEL + t * 16 + lr] = acc[t][r] * rl;
    }
}

// ---------------------------------------------------------------------------
// Kernel 3: out = y @ w_out + b_out.  Y flattened [16384 x 768].
// grid (12, 128), block 256. Same async-LDS W staging as kernel 1.
// ---------------------------------------------------------------------------
__global__ __launch_bounds__(256)
void out_gemm_kernel(const float* __restrict__ yin, const float* __restrict__ w,
                     const float* __restrict__ bias, float* __restrict__ out) {
    __shared__ float wbuf[2][16 * 72];

    const int tid  = threadIdx.x;
    const int wave = tid >> 5;
    const int lane = tid & 31;
    const int lr   = lane & 15;
    const int half = lane >> 4;

    const int m0   = blockIdx.y * 128 + wave * 16;
    const int col0 = blockIdx.x * 64;

    const int srow_ld = tid >> 4;
    const int scol_ld = (tid & 15) * 4;

    const float* A = yin + (size_t)(m0 + lr) * D_MODEL + 2 * half;

    async_load_b128(lds_off_of(&wbuf[0][srow_ld * 72 + scol_ld]),
                    w + (size_t)srow_ld * D_MODEL + col0 + scol_ld);

    v8f acc[4] = {};
    const int NK = D_MODEL / 16;           // 48
    for (int kt = 0; kt < NK; ++kt) {
        wait_async0();
        __syncthreads();
        if (kt + 1 < NK) {
            async_load_b128(lds_off_of(&wbuf[(kt + 1) & 1][srow_ld * 72 + scol_ld]),
                            w + (size_t)((kt + 1) * 16 + srow_ld) * D_MODEL + col0 + scol_ld);
        }
        const float* wt = &wbuf[kt & 1][0];
#pragma unroll
        for (int k0s = 0; k0s < 4; ++k0s) {
            v2f a = *(const v2f*)(A + kt * 16 + k0s * 4);
            int krow = k0s * 4 + 2 * half;
#pragma unroll
            for (int t = 0; t < 4; ++t) {
                v2f b;
                b.x = wt[krow * 72 + t * 16 + lr];
                b.y = wt[(krow + 1) * 72 + t * 16 + lr];
                acc[t] = wmma4(a, b, acc[t]);
            }
        }
    }

#pragma unroll
    for (int t = 0; t < 4; ++t) {
        int col  = col0 + t * 16 + lr;
        float bv = bias[col];
#pragma unroll
        for (int r = 0; r < 8; ++r) {
            int row = m0 + r + 8 * half;
            out[(size_t)row * D_MODEL + col] = acc[t][r] + bv;
        }
    }
}

// ---------------------------------------------------------------------------
extern "C" void kernel_launch(void* const* d_in, const int* in_sizes, int n_in,
                              void* d_out, int out_size, void* d_ws, size_t ws_size,
                              hipStream_t stream) {
    const float* x      = (const float*)d_in[0];
    const float* w_qkv  = (const float*)d_in[1];
    const float* b_qkv  = (const float*)d_in[2];
    const float* w_out  = (const float*)d_in[3];
    const float* b_out  = (const float*)d_in[4];
    const unsigned char* attn_mask = (const unsigned char*)d_in[5];

    float* ws = (float*)d_ws;          // [Q | K | V | Y], each QKVSZ floats (Y = NSEQ*S*D)
    float* y  = ws + 3 * QKVSZ;
    float* out = (float*)d_out;

    qkv_gemm_kernel<<<dim3(D3 / 64, S_LEN / 128, NSEQ), 256, 0, stream>>>(x, w_qkv, b_qkv, ws);
    attn_kernel<<<dim3(S_LEN / 128, NHEAD, NSEQ), 256, 0, stream>>>(ws, attn_mask, y);
    out_gemm_kernel<<<dim3(D_MODEL / 64, (NSEQ * S_LEN) / 128), 256, 0, stream>>>(y, w_out, b_out, out);
}